// TripletLossCosine_77781857731082
// MI455X (gfx1250) — compile-verified
//
#include <hip/hip_runtime.h>
#include <math.h>

#define NROWS 8192
#define DDIM  256

typedef __attribute__((ext_vector_type(2))) float v2f;
typedef __attribute__((ext_vector_type(8))) float v8f;

// Exact (no-rounding) float max via integer atomics: order-independent => deterministic.
__device__ __forceinline__ void atomicMaxFloat(float* addr, float val) {
    if (val >= 0.0f) {
        atomicMax((int*)addr, __float_as_int(val));
    } else {
        atomicMin((unsigned int*)addr, __float_as_uint(val));
    }
}

__device__ __forceinline__ float vmax8(v8f c) {
    return fmaxf(fmaxf(fmaxf(c[0], c[1]), fmaxf(c[2], c[3])),
                 fmaxf(fmaxf(c[4], c[5]), fmaxf(c[6], c[7])));
}

__global__ __launch_bounds__(256) void init_maxq_kernel(float* __restrict__ max_q) {
    int i = blockIdx.x * 256 + threadIdx.x;
    if (i < NROWS) max_q[i] = -INFINITY;
}

// Fused scores-GEMM + column-max using V_WMMA_F32_16X16X4_F32.
// grid = (NROWS/512 col blocks, NROWS/256 row chunks), block = 256 threads = 8 waves.
// Each wave: 64 columns x 32 rows register tile = 2x4 = 8 independent WMMA chains,
// so each k-step is 6 global_load_b64 feeding 8 v_wmma (0.75 loads/WMMA).
__global__ __launch_bounds__(256) void colmax_wmma_kernel(const float* __restrict__ m,
                                                          const float* __restrict__ q,
                                                          float* __restrict__ max_q) {
    const int lane   = threadIdx.x & 31;
    const int wave   = threadIdx.x >> 5;
    const int colBase = blockIdx.x * 512 + wave * 64;   // this wave's 64 columns
    const int rowBase = blockIdx.y * 256;               // this block's 256 rows
    const int r    = lane & 15;                         // M for A-tile, N (column) for B-tile
    const int kOff = (lane >> 4) * 2;                   // half-wave K offset (ISA 16x4 layout)

    // B tile base pointer for the 4 column groups (layout identical to A-tile loads)
    const float* qb = q + (size_t)(colBase + r) * DDIM + kOff;

    float cmax0 = -INFINITY, cmax1 = -INFINITY, cmax2 = -INFINITY, cmax3 = -INFINITY;

    for (int mt = 0; mt < 8; ++mt) {
        const float* ab0 = m + (size_t)(rowBase + mt * 32 + r) * DDIM + kOff;
        const float* ab1 = ab0 + 16 * DDIM;
        v8f c00 = {}, c01 = {}, c02 = {}, c03 = {};
        v8f c10 = {}, c11 = {}, c12 = {}, c13 = {};
#pragma unroll 2
        for (int k = 0; k < DDIM; k += 4) {
            v2f a0 = *(const v2f*)(ab0 + k);
            v2f a1 = *(const v2f*)(ab1 + k);
            v2f b0 = *(const v2f*)(qb + k);
            v2f b1 = *(const v2f*)(qb + 16 * DDIM + k);
            v2f b2 = *(const v2f*)(qb + 32 * DDIM + k);
            v2f b3 = *(const v2f*)(qb + 48 * DDIM + k);
            // 8 args: (neg_a, A, neg_b, B, c_mod, C, reuse_a, reuse_b)
            c00 = __builtin_amdgcn_wmma_f32_16x16x4_f32(false, a0, false, b0, (short)0, c00, false, false);
            c01 = __builtin_amdgcn_wmma_f32_16x16x4_f32(false, a0, false, b1, (short)0, c01, false, false);
            c02 = __builtin_amdgcn_wmma_f32_16x16x4_f32(false, a0, false, b2, (short)0, c02, false, false);
            c03 = __builtin_amdgcn_wmma_f32_16x16x4_f32(false, a0, false, b3, (short)0, c03, false, false);
            c10 = __builtin_amdgcn_wmma_f32_16x16x4_f32(false, a1, false, b0, (short)0, c10, false, false);
            c11 = __builtin_amdgcn_wmma_f32_16x16x4_f32(false, a1, false, b1, (short)0, c11, false, false);
            c12 = __builtin_amdgcn_wmma_f32_16x16x4_f32(false, a1, false, b2, (short)0, c12, false, false);
            c13 = __builtin_amdgcn_wmma_f32_16x16x4_f32(false, a1, false, b3, (short)0, c13, false, false);
        }
        // Fold the 16 rows held per lane (both M-tiles) into the running per-lane column max.
        cmax0 = fmaxf(cmax0, fmaxf(vmax8(c00), vmax8(c10)));
        cmax1 = fmaxf(cmax1, fmaxf(vmax8(c01), vmax8(c11)));
        cmax2 = fmaxf(cmax2, fmaxf(vmax8(c02), vmax8(c12)));
        cmax3 = fmaxf(cmax3, fmaxf(vmax8(c03), vmax8(c13)));
    }

    // Merge the two half-wave M-ranges (lanes L and L+16 hold the same column N=L%16).
    cmax0 = fmaxf(cmax0, __shfl_xor(cmax0, 16, 32));
    cmax1 = fmaxf(cmax1, __shfl_xor(cmax1, 16, 32));
    cmax2 = fmaxf(cmax2, __shfl_xor(cmax2, 16, 32));
    cmax3 = fmaxf(cmax3, __shfl_xor(cmax3, 16, 32));

    if (lane < 16) {
        atomicMaxFloat(&max_q[colBase +  0 + lane], cmax0);
        atomicMaxFloat(&max_q[colBase + 16 + lane], cmax1);
        atomicMaxFloat(&max_q[colBase + 32 + lane], cmax2);
        atomicMaxFloat(&max_q[colBase + 48 + lane], cmax3);
    }
}

// diag[j] = <m_j, q_j>; one wave per column, fixed-order shuffle reduce (deterministic).
__global__ __launch_bounds__(256) void diag_kernel(const float* __restrict__ m,
                                                   const float* __restrict__ q,
                                                   float* __restrict__ diag) {
    const int lane = threadIdx.x & 31;
    const int wave = threadIdx.x >> 5;
    const int j = blockIdx.x * 8 + wave;
    const float* mr = m + (size_t)j * DDIM;
    const float* qr = q + (size_t)j * DDIM;
    float s = 0.0f;
#pragma unroll
    for (int k = lane * 4; k < DDIM; k += 32 * 4) {
        float4 a = *(const float4*)(mr + k);
        float4 b = *(const float4*)(qr + k);
        s += a.x * b.x + a.y * b.y + a.z * b.z + a.w * b.w;
    }
    for (int off = 16; off > 0; off >>= 1) s += __shfl_down(s, off, 32);
    if (lane == 0) diag[j] = s;
}

// Final hinge-loss mean: single block, fixed-order LDS tree => deterministic.
__global__ __launch_bounds__(256) void loss_kernel(const float* __restrict__ max_q,
                                                   const float* __restrict__ diag,
                                                   const float* __restrict__ labels,
                                                   float* __restrict__ out) {
    __shared__ float sm[256];
    float s = 0.0f;
    for (int j = threadIdx.x; j < NROWS; j += 256) {
        float mq  = max_q[j];
        float tar = (mq == labels[j]) ? 1.0f : -1.0f;
        float v   = -tar * (mq - diag[j]);
        s += fmaxf(v, 0.0f);
    }
    sm[threadIdx.x] = s;
    __syncthreads();
    for (int off = 128; off > 0; off >>= 1) {
        if ((int)threadIdx.x < off) sm[threadIdx.x] += sm[threadIdx.x + off];
        __syncthreads();
    }
    if (threadIdx.x == 0) out[0] = sm[0] / (float)NROWS;
}

extern "C" void kernel_launch(void* const* d_in, const int* in_sizes, int n_in,
                              void* d_out, int out_size, void* d_ws, size_t ws_size,
                              hipStream_t stream) {
    const float* m      = (const float*)d_in[0];
    const float* q      = (const float*)d_in[1];
    const float* labels = (const float*)d_in[2];
    float* out   = (float*)d_out;
    float* max_q = (float*)d_ws;          // NROWS floats
    float* diag  = max_q + NROWS;         // NROWS floats (64 KB total)

    init_maxq_kernel<<<NROWS / 256, 256, 0, stream>>>(max_q);
    colmax_wmma_kernel<<<dim3(NROWS / 512, NROWS / 256), 256, 0, stream>>>(m, q, max_q);
    diag_kernel<<<NROWS / 8, 256, 0, stream>>>(m, q, diag);
    loss_kernel<<<1, 256, 0, stream>>>(max_q, diag, labels, out);
}